// DotProductKernel_4063039062479
// MI455X (gfx1250) — compile-verified
//
#include <hip/hip_runtime.h>

typedef __attribute__((ext_vector_type(2))) float v2f;
typedef __attribute__((ext_vector_type(4))) float v4f;
typedef __attribute__((ext_vector_type(8))) float v8f;

#define KDIM   256   // input feature dim (k_dim == q_dim)
#define EDIM   512   // embed dim
#define NHEAD  8
#define HDIM   64    // EDIM / NHEAD
#define MNBR   32    // neighbors per query (== wave32 lanes)
#define NSEQ   4096  // sequence length per batch

// ---------------------------------------------------------------------------
// Kernel 1: fp32 projection GEMM  Y[r, c] = sum_k X[r, k] * W[c, k] + bias[c]
//   X: (rows x 256) row-major, W: (512 x 256) row-major (torch Linear weight),
//   Y: (rows x 512) row-major.
// One wave computes one 16x16 tile of Y via V_WMMA_F32_16X16X4_F32,
// stepping K by 4 (64 WMMAs per tile).
//
// fp32 A fragment (16x4, M x K), per ISA layout:
//   lanes 0-15 : M = lane,    {v0,v1} = K = kk+0, kk+1
//   lanes 16-31: M = lane-16, {v0,v1} = K = kk+2, kk+3
// fp32 B fragment (4x16, K x N):
//   lanes 0-15 : N = lane,    {v0,v1} = K rows kk+0, kk+1
//   lanes 16-31: N = lane-16, {v0,v1} = K rows kk+2, kk+3
// Both fragments are a float2 load at (row)*256 + kk + 2*(lane>=16).
// ---------------------------------------------------------------------------
__global__ __launch_bounds__(128) void proj_wmma(
    const float* __restrict__ X, const float* __restrict__ W,
    const float* __restrict__ bias, float* __restrict__ Y) {
  const int lane = threadIdx.x & 31;
  const int wave = threadIdx.x >> 5;          // 4 waves per block
  const int l    = lane & 15;
  const int half = lane >> 4;                 // 0: lanes 0-15, 1: lanes 16-31

  const int row0 = blockIdx.x * 16;                    // output row tile
  const int col0 = (blockIdx.y * 4 + wave) * 16;       // output col tile

  const float* __restrict__ aPtr = X + (size_t)(row0 + l) * KDIM + 2 * half;
  const float* __restrict__ bPtr = W + (size_t)(col0 + l) * KDIM + 2 * half;

  v8f acc = {};
#pragma unroll 8
  for (int kk = 0; kk < KDIM; kk += 4) {
    v2f a = *(const v2f*)(aPtr + kk);
    v2f b = *(const v2f*)(bPtr + kk);
    // D = A(16x4) * B(4x16) + C  -- fp32 in, fp32 accumulate
    acc = __builtin_amdgcn_wmma_f32_16x16x4_f32(
        /*neg_a=*/false, a, /*neg_b=*/false, b,
        /*c_mod=*/(short)0, acc, /*reuse_a=*/false, /*reuse_b=*/false);
  }

  // C/D layout: VGPR r -> row (r + 8*half), column = l. Add bias, store.
  const float bv = bias[col0 + l];
#pragma unroll
  for (int r = 0; r < 8; ++r) {
    const int mrow = r + 8 * half;
    Y[(size_t)(row0 + mrow) * EDIM + col0 + l] = acc[r] + bv;
  }
}

// ---------------------------------------------------------------------------
// Kernel 2: gathered per-head dot products.
//   out[b,i,j,h] = 0.125 * sum_d Q[b,i,h*64+d] * K[b, idx[b,i,j], h*64+d]
// One wave per query (b,i); lane j handles neighbor j (m == 32 == wave32).
// Q-row loads are wave-uniform (broadcast from cache); K rows are random
// 2KB gathers that hit the 192MB L2 (whole K buffer is 16MB).
// ---------------------------------------------------------------------------
__global__ __launch_bounds__(256) void gather_scores(
    const float* __restrict__ Qbuf, const float* __restrict__ Kbuf,
    const long long* __restrict__ nbhd, float* __restrict__ out) {
  const int lane  = threadIdx.x & 31;
  const int wib   = threadIdx.x >> 5;                       // wave in block
  const int qg    = blockIdx.x * (blockDim.x >> 5) + wib;   // global query 0..8191
  const int batch = qg / NSEQ;

  const long long nb = nbhd[(size_t)qg * MNBR + lane];
  const float* __restrict__ Krow = Kbuf + ((size_t)batch * NSEQ + (size_t)nb) * EDIM;
  const float* __restrict__ Qrow = Qbuf + (size_t)qg * EDIM;

  v4f accLo = {}, accHi = {};
#pragma unroll
  for (int h = 0; h < NHEAD; ++h) {
    float s = 0.0f;
#pragma unroll
    for (int d = 0; d < HDIM; d += 4) {
      v4f qv = *(const v4f*)(Qrow + h * HDIM + d);   // uniform -> broadcast
      v4f kv = *(const v4f*)(Krow + h * HDIM + d);   // b128 gather
      s = __builtin_fmaf(qv.x, kv.x, s);
      s = __builtin_fmaf(qv.y, kv.y, s);
      s = __builtin_fmaf(qv.z, kv.z, s);
      s = __builtin_fmaf(qv.w, kv.w, s);
    }
    s *= 0.125f;  // 1/sqrt(head_dim=64)
    if (h < 4) accLo[h] = s; else accHi[h - 4] = s;
  }

  float* o = out + ((size_t)qg * MNBR + lane) * NHEAD;
  *(v4f*)(o)     = accLo;
  *(v4f*)(o + 4) = accHi;
}

extern "C" void kernel_launch(void* const* d_in, const int* in_sizes, int n_in,
                              void* d_out, int out_size, void* d_ws, size_t ws_size,
                              hipStream_t stream) {
  const float*     k    = (const float*)d_in[0];      // (b, n, 256)
  const float*     q    = (const float*)d_in[1];      // (b, n, 256)
  const long long* nb   = (const long long*)d_in[2];  // (b, n, 32) int64
  const float*     Wk   = (const float*)d_in[3];      // (512, 256)
  const float*     bk   = (const float*)d_in[4];      // (512,)
  const float*     Wq   = (const float*)d_in[5];      // (512, 256)
  const float*     bq   = (const float*)d_in[6];      // (512,)
  float*           out  = (float*)d_out;              // (b, n, 32, 8)

  const int bn = in_sizes[0] / KDIM;                  // b*n = 8192 rows

  float* Kbuf = (float*)d_ws;                         // bn*512 floats = 16 MB
  float* Qbuf = Kbuf + (size_t)bn * EDIM;             // next 16 MB

  // Projections: grid (row tiles, col tiles / 4 waves), 4 waves per block.
  dim3 blockP(128, 1, 1);
  dim3 gridP(bn / 16, EDIM / (16 * 4), 1);            // (512, 8)
  proj_wmma<<<gridP, blockP, 0, stream>>>(k, Wk, bk, Kbuf);
  proj_wmma<<<gridP, blockP, 0, stream>>>(q, Wq, bq, Qbuf);

  // Gathered scores: one wave per query, 8 waves per block.
  dim3 blockG(256, 1, 1);
  dim3 gridG(bn / 8, 1, 1);
  gather_scores<<<gridG, blockG, 0, stream>>>(Qbuf, Kbuf, nb, out);
}